// TimbreAttention_68118181314791
// MI455X (gfx1250) — compile-verified
//
#include <hip/hip_runtime.h>

typedef __attribute__((ext_vector_type(2))) float v2f;
typedef __attribute__((ext_vector_type(8))) float v8f;

#define BB 8
#define CC 2
#define FF 352
#define LL 2048
#define NN (FF*LL)        /* 720896 tokens per batch, < 2^20 */
#define EMB 6
#define KTOP 84
#define HBINS 2048        /* max bins (11-bit level); hist stride */

__device__ __forceinline__ int imin(int a, int b) { return a < b ? a : b; }

__device__ __forceinline__ unsigned int fkey(float s) {
  unsigned int u = __float_as_uint(s);
  return (u & 0x80000000u) ? ~u : (u | 0x80000000u);  // monotonic: bigger float -> bigger key
}
// 52-bit unique key: [51:20]=sortable score bits, [19:0]=NN-1-n.
// Among equal scores, smaller index -> larger key (matches jax top_k tie-break).
__device__ __forceinline__ unsigned long long key52(float s, int n) {
  return ((unsigned long long)fkey(s) << 20) | (unsigned int)(NN - 1 - n);
}

// ---------------- init workspace state ----------------
__global__ void k_init(int* hist, unsigned long long* prefix, int* kRem, int* selCnt,
                       float* selScore, int* selIdx) {
  int t = blockIdx.x * blockDim.x + threadIdx.x;
  if (t < BB * HBINS) hist[t] = 0;
  if (t < BB) { prefix[t] = 0ull; kRem[t] = KTOP; selCnt[t] = 0; }
  if (t < BB * KTOP) { selScore[t] = 0.f; selIdx[t] = 0; }
}

// ---------------- per-row stats: sum, x[0], x[1], x[L-1] ----------------
__global__ void k_rowstats(const float* __restrict__ x, float4* __restrict__ rs) {
  int row = blockIdx.x;                       // (b*CC + c)*FF + fi, 0..5631
  const float* p = x + (size_t)row * LL;
  float s = 0.f;
  for (int i = threadIdx.x; i < LL; i += blockDim.x) s += p[i];
  __shared__ float red[256];
  red[threadIdx.x] = s; __syncthreads();
  for (int off = 128; off > 0; off >>= 1) {
    if (threadIdx.x < off) red[threadIdx.x] += red[threadIdx.x + off];
    __syncthreads();
  }
  if (threadIdx.x == 0) rs[row] = make_float4(red[0], p[0], p[1], p[LL - 1]);
}

// ---------------- query[b][6] ----------------
__global__ void k_query(const float4* __restrict__ rs, float* __restrict__ q) {
  int bc = blockIdx.x;                         // 0..15
  int b = bc / CC, c = bc % CC;
  float sS = 0.f, s0 = 0.f, s1 = 0.f, sL = 0.f;
  for (int fi = threadIdx.x; fi < FF; fi += blockDim.x) {
    float4 v = rs[bc * FF + fi];
    sS += v.x; s0 += v.y; s1 += v.z; sL += v.w;
  }
  __shared__ float rA[128], rB[128], rC[128], rD[128];
  rA[threadIdx.x] = sS; rB[threadIdx.x] = s0; rC[threadIdx.x] = s1; rD[threadIdx.x] = sL;
  __syncthreads();
  for (int off = 64; off > 0; off >>= 1) {
    if (threadIdx.x < off) {
      rA[threadIdx.x] += rA[threadIdx.x + off];
      rB[threadIdx.x] += rB[threadIdx.x + off];
      rC[threadIdx.x] += rC[threadIdx.x + off];
      rD[threadIdx.x] += rD[threadIdx.x + off];
    }
    __syncthreads();
  }
  if (threadIdx.x == 0) {
    float inv = 1.0f / (float)NN;
    float S = rA[0], p0 = rB[0], p1 = rC[0], xl = rD[0];
    q[b * EMB + 0 + c] = S * inv;                              // shift 0
    q[b * EMB + 2 + c] = (S - p0 + xl) * inv;                  // shift 1
    q[b * EMB + 4 + c] = (S - (p0 + p1) + 2.f * xl) * inv;     // shift 2
  }
}

// ---------------- scores via V_WMMA_F32_16X16X4_F32, fused radix level 0 ----------------
// Per wave: 16 tokens. A(16x4)=emb dims e0..e3 (ISA layout: lanes16-31 hold K=2,3),
// B(4x16)=query broadcast across columns. Second WMMA accumulates e4,e5 (K=2,3 zero).
// D[m][n] = score(token m) replicated over n; lanes 0 and 16 write 8 scores each and
// also bin them into the level-0 (bits 51:41, 2048-bin) histogram.
__global__ void k_scores(const float* __restrict__ x, const float* __restrict__ q,
                         float* __restrict__ sc, int* __restrict__ hist) {
  __shared__ int lh[HBINS];
  for (int i = threadIdx.x; i < HBINS; i += blockDim.x) lh[i] = 0;
  __syncthreads();

  int b = blockIdx.y;
  int lane = threadIdx.x & 31;
  int wave = threadIdx.x >> 5;                 // 0..7
  int g = blockIdx.x * 8 + wave;               // 16-token group id, 0..45055
  int m = lane & 15;
  int token = g * 16 + m;
  int fi = token >> 11, li = token & (LL - 1);
  const float* x0 = x + (size_t)b * (CC * FF * LL) + (size_t)fi * LL; // c=0 row
  const float* x1 = x0 + (size_t)(FF * LL);                          // c=1 row
  const float* qb = q + b * EMB;
  bool hi = lane >= 16;

  int liA = imin(li + (hi ? 1 : 0), LL - 1);
  v2f a, bm, a2, b2;
  a.x = x0[liA];                 // lo: e0=x[c0,li]    hi: e2=x[c0,li+1]
  a.y = x1[liA];                 // lo: e1=x[c1,li]    hi: e3=x[c1,li+1]
  bm.x = hi ? qb[2] : qb[0];
  bm.y = hi ? qb[3] : qb[1];

  int liB = imin(li + 2, LL - 1);
  a2.x = hi ? 0.f : x0[liB];     // K=0: e4 ; hi lanes (K=2,3) zero
  a2.y = hi ? 0.f : x1[liB];     // K=1: e5
  b2.x = hi ? 0.f : qb[4];
  b2.y = hi ? 0.f : qb[5];

  v8f cacc = {};
  cacc = __builtin_amdgcn_wmma_f32_16x16x4_f32(false, a,  false, bm, (short)0, cacc, false, false);
  cacc = __builtin_amdgcn_wmma_f32_16x16x4_f32(false, a2, false, b2, (short)0, cacc, false, false);

  union { v8f v; float f[8]; } d; d.v = cacc;
  if ((lane & 15) == 0) {        // lane 0 -> tokens m=0..7, lane 16 -> m=8..15
    int base = g * 16 + (hi ? 8 : 0);
    float* out = sc + (size_t)b * NN + base;
    #pragma unroll
    for (int r = 0; r < 8; ++r) {
      out[r] = d.f[r];
      unsigned long long k = key52(d.f[r], base + r);
      atomicAdd(&lh[(int)(k >> 41)], 1);       // level 0: top 11 bits
    }
  }
  __syncthreads();
  for (int i = threadIdx.x; i < HBINS; i += blockDim.x) {
    int v = lh[i];
    if (v) atomicAdd(&hist[b * HBINS + i], v);
  }
}

// ---------------- radix-select refinement levels ----------------
__global__ void k_hist(const float* __restrict__ sc, const unsigned long long* __restrict__ prefix,
                       int* __restrict__ hist, int shift, int width) {
  int b = blockIdx.y;
  int bins = 1 << width;
  __shared__ int lh[HBINS];
  for (int i = threadIdx.x; i < bins; i += blockDim.x) lh[i] = 0;
  __syncthreads();
  unsigned long long pref = prefix[b];
  int chkShift = shift + width;
  const float* s = sc + (size_t)b * NN;
  for (int n = blockIdx.x * blockDim.x + threadIdx.x; n < NN; n += gridDim.x * blockDim.x) {
    unsigned long long k = key52(s[n], n);
    if ((k >> chkShift) == pref)
      atomicAdd(&lh[(int)((k >> shift) & (bins - 1))], 1);
  }
  __syncthreads();
  for (int i = threadIdx.x; i < bins; i += blockDim.x) {
    int v = lh[i];
    if (v) atomicAdd(&hist[b * HBINS + i], v);
  }
}

__global__ void k_scan(int* hist, unsigned long long* prefix, int* kRem, int width) {
  int t = threadIdx.x;
  if (t < BB) {
    int bins = 1 << width;
    int need = kRem[t], cnt = 0, chosen = 0;
    for (int bin = bins - 1; bin >= 0; --bin) {
      int c = hist[t * HBINS + bin];
      if (cnt + c >= need) { chosen = bin; break; }
      cnt += c;
    }
    prefix[t] = (prefix[t] << width) | (unsigned int)chosen;
    kRem[t] = need - cnt;
  }
  __syncthreads();
  for (int i = threadIdx.x; i < BB * HBINS; i += blockDim.x) hist[i] = 0;  // ready for next level
}

__global__ void k_gather(const float* __restrict__ sc, const unsigned long long* __restrict__ prefix,
                         int* selCnt, float* selScore, int* selIdx) {
  int b = blockIdx.y;
  unsigned long long thr = prefix[b];          // exact 52-bit key of 84th-largest; keys unique
  const float* s = sc + (size_t)b * NN;
  for (int n = blockIdx.x * blockDim.x + threadIdx.x; n < NN; n += gridDim.x * blockDim.x) {
    if (key52(s[n], n) >= thr) {
      int slot = atomicAdd(&selCnt[b], 1);
      if (slot < KTOP) { selScore[b * KTOP + slot] = s[n]; selIdx[b * KTOP + slot] = n; }
    }
  }
}

// ---------------- softmax + weighted gather + blend ----------------
__global__ void k_final(const float* __restrict__ x, const float* __restrict__ q,
                        const float* __restrict__ selScore, const int* __restrict__ selIdx,
                        const float* __restrict__ wptr, float* __restrict__ out) {
  int b = blockIdx.x;
  __shared__ float attn[KTOP];
  __shared__ int idx[KTOP];
  if (threadIdx.x < KTOP) {
    attn[threadIdx.x] = selScore[b * KTOP + threadIdx.x];
    idx[threadIdx.x]  = selIdx[b * KTOP + threadIdx.x];
  }
  __syncthreads();
  if (threadIdx.x == 0) {
    // deterministic order: insertion-sort by token index
    for (int i = 1; i < KTOP; ++i) {
      float sv = attn[i]; int iv = idx[i]; int j = i - 1;
      while (j >= 0 && idx[j] > iv) { idx[j + 1] = idx[j]; attn[j + 1] = attn[j]; --j; }
      idx[j + 1] = iv; attn[j + 1] = sv;
    }
    float mx = attn[0];
    for (int k = 1; k < KTOP; ++k) mx = fmaxf(mx, attn[k]);
    float sum = 0.f;
    for (int k = 0; k < KTOP; ++k) { attn[k] = __expf(attn[k] - mx); sum += attn[k]; }
    float inv = 1.f / sum;
    for (int k = 0; k < KTOP; ++k) attn[k] *= inv;
  }
  __syncthreads();
  float w = *wptr;
  if (threadIdx.x < EMB) {
    int e = threadIdx.x;
    int c = e & 1, sh = e >> 1;                // e = 2*shift + channel
    const float* xr = x + (size_t)b * (CC * FF * LL) + (size_t)c * (FF * LL);
    float acc = 0.f;
    for (int k = 0; k < KTOP; ++k) {
      int t = idx[k];
      int fi = t >> 11, li = t & (LL - 1);
      acc += attn[k] * xr[(size_t)fi * LL + imin(li + sh, LL - 1)];
    }
    out[b * EMB + e] = acc * w + q[b * EMB + e] * (0.5f - w);
  }
}

extern "C" void kernel_launch(void* const* d_in, const int* in_sizes, int n_in,
                              void* d_out, int out_size, void* d_ws, size_t ws_size,
                              hipStream_t stream) {
  const float* x = (const float*)d_in[0];
  const float* w = (const float*)d_in[1];
  float* out = (float*)d_out;

  char* ws = (char*)d_ws;
  size_t off = 0;
  float* sc = (float*)(ws + off);                  off += (size_t)BB * NN * sizeof(float);
  float4* rs = (float4*)(ws + off);                off += (size_t)BB * CC * FF * sizeof(float4);
  float* q = (float*)(ws + off);                   off += 64 * sizeof(float);
  int* hist = (int*)(ws + off);                    off += (size_t)BB * HBINS * sizeof(int);
  unsigned long long* prefix = (unsigned long long*)(ws + off); off += BB * sizeof(unsigned long long);
  int* kRem = (int*)(ws + off);                    off += 16 * sizeof(int);
  int* selCnt = (int*)(ws + off);                  off += 16 * sizeof(int);
  float* selScore = (float*)(ws + off);            off += (size_t)BB * KTOP * sizeof(float);
  int* selIdx = (int*)(ws + off);                  off += (size_t)BB * KTOP * sizeof(int);
  (void)off; (void)ws_size; (void)in_sizes; (void)n_in; (void)out_size;

  // 52-bit key split into 5 radix levels (level 0 fused into k_scores)
  static const int shifts[5] = {41, 30, 20, 10, 0};
  static const int widths[5] = {11, 11, 10, 10, 10};

  k_init<<<(BB * HBINS + 255) / 256, 256, 0, stream>>>(hist, prefix, kRem, selCnt, selScore, selIdx);
  k_rowstats<<<BB * CC * FF, 256, 0, stream>>>(x, rs);
  k_query<<<BB * CC, 128, 0, stream>>>(rs, q);
  k_scores<<<dim3(NN / 16 / 8, BB), 256, 0, stream>>>(x, q, sc, hist);   // 5632 x 8 blocks
  k_scan<<<1, 256, 0, stream>>>(hist, prefix, kRem, widths[0]);
  for (int p = 1; p < 5; ++p) {
    k_hist<<<dim3(1024, BB), 256, 0, stream>>>(sc, prefix, hist, shifts[p], widths[p]);
    k_scan<<<1, 256, 0, stream>>>(hist, prefix, kRem, widths[p]);
  }
  k_gather<<<dim3(1024, BB), 256, 0, stream>>>(sc, prefix, selCnt, selScore, selIdx);
  k_final<<<BB, 128, 0, stream>>>(x, q, selScore, selIdx, w, out);
}